// DANet3D_20607253086547
// MI455X (gfx1250) — compile-verified
//
#include <hip/hip_runtime.h>
#include <math.h>

// Problem constants (DANet3D): B=2, C=64, INTER=32, N = 20*20*20 = 8000
#define BB 2
#define CC 64
#define II 32
#define NN 8000

typedef __attribute__((ext_vector_type(2))) float v2f;
typedef __attribute__((ext_vector_type(8))) float v8f;

// D = A(16x4) * B(4x16) + C(16x16), all f32. 8-arg VOP3P form:
// (neg_a, A, neg_b, B, c_mod, C, reuse_a, reuse_b)
__device__ __forceinline__ v8f wmma4(v2f a, v2f b, v8f c) {
  return __builtin_amdgcn_wmma_f32_16x16x4_f32(false, a, false, b, (short)0, c,
                                               false, false);
}

// ---------------------------------------------------------------------------
// Kernel 1: QKV projections.
//   q[b][i][n] = Wq[i,:]·x[b,:,n] + bq[i]   (i-major so flash A/B loads coalesce)
//   k[b][i][n] = Wk[i,:]·x[b,:,n] + bk[i]
//   vT[b][n][c] = Wv[c,:]·x[b,:,n] + bv[c]  (n-major so PV B-operand coalesces)
// ---------------------------------------------------------------------------
__global__ void qkv_kernel(const float* __restrict__ x,
                           const float* __restrict__ Wq, const float* __restrict__ bq,
                           const float* __restrict__ Wk, const float* __restrict__ bk,
                           const float* __restrict__ Wv, const float* __restrict__ bv,
                           float* __restrict__ q, float* __restrict__ k,
                           float* __restrict__ vT) {
  int idx = blockIdx.x * blockDim.x + threadIdx.x;
  if (idx >= BB * NN) return;
  int b = idx / NN;
  int n = idx - b * NN;
  const float* xb = x + (size_t)b * CC * NN + n;

  float xv[CC];
#pragma unroll
  for (int c = 0; c < CC; ++c) xv[c] = xb[(size_t)c * NN];

  for (int i = 0; i < II; ++i) {
    float aq = bq[i], ak = bk[i];
    for (int c = 0; c < CC; ++c) {
      aq = fmaf(Wq[i * CC + c], xv[c], aq);
      ak = fmaf(Wk[i * CC + c], xv[c], ak);
    }
    q[(size_t)b * II * NN + (size_t)i * NN + n] = aq;
    k[(size_t)b * II * NN + (size_t)i * NN + n] = ak;
  }
  for (int o = 0; o < CC; ++o) {
    float av = bv[o];
    for (int c = 0; c < CC; ++c) av = fmaf(Wv[o * CC + c], xv[c], av);
    vT[(size_t)b * NN * CC + (size_t)n * CC + o] = av;
  }
}

// ---------------------------------------------------------------------------
// Kernel 2: channel-attention energy E_c[b] = P P^T  (64x64, K=8000).
// One wave per 16x16 tile; grid = B*4*4 tiles, block = 32.
// ---------------------------------------------------------------------------
__global__ void energy_c_kernel(const float* __restrict__ x,
                                float* __restrict__ ec) {
  int tile = blockIdx.x;
  int b = tile >> 4;
  int rb = (tile >> 2) & 3;
  int cb = tile & 3;
  const float* proj = x + (size_t)b * CC * NN;

  int lane = threadIdx.x & 31;
  int lm = lane & 15;
  int koff = (lane >> 4) * 2;  // lanes 0-15: K 0..1, lanes 16-31: K 2..3

  const float* arow = proj + (size_t)(rb * 16 + lm) * NN + koff;
  const float* brow = proj + (size_t)(cb * 16 + lm) * NN + koff;

  v8f acc = {};
  for (int kk = 0; kk < NN; kk += 4) {
    v2f a = *(const v2f*)(arow + kk);
    v2f bvv = *(const v2f*)(brow + kk);
    acc = wmma4(a, bvv, acc);
  }
  float* dst = ec + b * CC * CC;
#pragma unroll
  for (int i = 0; i < 8; ++i) {
    int row = rb * 16 + i + 8 * (lane >> 4);
    dst[row * CC + cb * 16 + lm] = acc[i];
  }
}

// ---------------------------------------------------------------------------
// Kernel 3: row softmax on E_c (B*64 rows of 64). One thread per row.
// ---------------------------------------------------------------------------
__global__ void softmax_c_kernel(float* __restrict__ ec) {
  int row = blockIdx.x * blockDim.x + threadIdx.x;
  if (row >= BB * CC) return;
  float* r = ec + (size_t)row * CC;
  float mx = r[0];
  for (int j = 1; j < CC; ++j) mx = fmaxf(mx, r[j]);
  float s = 0.f;
  for (int j = 0; j < CC; ++j) {
    float e = __expf(r[j] - mx);
    r[j] = e;
    s += e;
  }
  float inv = 1.f / s;
  for (int j = 0; j < CC; ++j) r[j] *= inv;
}

// ---------------------------------------------------------------------------
// Kernel 4: out_c = A_c (64x64) * P (64xN). One wave per 16x16 output tile.
// grid = B * 4 * (N/16), block = 32.
// ---------------------------------------------------------------------------
__global__ void out_c_kernel(const float* __restrict__ x,
                             const float* __restrict__ attn,
                             float* __restrict__ outc) {
  int t = blockIdx.x;
  int b = t / (4 * (NN / 16));
  int r = t - b * (4 * (NN / 16));
  int rb = r / (NN / 16);
  int nb = r - rb * (NN / 16);
  int n0 = nb * 16;

  const float* proj = x + (size_t)b * CC * NN;
  int lane = threadIdx.x & 31;
  int lm = lane & 15;
  int koff = (lane >> 4) * 2;

  const float* arow = attn + b * CC * CC + (rb * 16 + lm) * CC + koff;

  v8f acc = {};
#pragma unroll
  for (int kk = 0; kk < CC; kk += 4) {
    v2f a = *(const v2f*)(arow + kk);
    v2f bvv = {proj[(size_t)(kk + koff) * NN + n0 + lm],
               proj[(size_t)(kk + koff + 1) * NN + n0 + lm]};
    acc = wmma4(a, bvv, acc);
  }
  float* dst = outc + (size_t)b * CC * NN;
#pragma unroll
  for (int i = 0; i < 8; ++i) {
    int row = rb * 16 + i + 8 * (lane >> 4);
    dst[(size_t)row * NN + n0 + lm] = acc[i];
  }
}

// ---------------------------------------------------------------------------
// Kernel 5: flash position attention.
//   E_p[n][m] = sum_i q[i][n]*k[i][m]; P = softmax_m; O[n][c] = sum_m P[n][m]*vT[m][c]
// One wave per 16-query block; streaming over all m in 16-tiles with online
// softmax. P is relaid out C->A via LDS (1 KB). grid = B * N/16, block = 32.
// ---------------------------------------------------------------------------
__global__ void pos_attn_kernel(const float* __restrict__ q,
                                const float* __restrict__ k,
                                const float* __restrict__ vT,
                                float* __restrict__ outpT) {
  __shared__ float pl[16 * 16];

  int t = blockIdx.x;
  int b = t / (NN / 16);
  int nb = t - b * (NN / 16);
  int n0 = nb * 16;

  const float* qb = q + (size_t)b * II * NN;
  const float* kb = k + (size_t)b * II * NN;
  const float* vb = vT + (size_t)b * NN * CC;

  int lane = threadIdx.x & 31;
  int lm = lane & 15;
  int hi = lane >> 4;
  int koff = hi * 2;

  // Q A-tiles for the 8 K-steps of the S GEMM (row = query = lm, K = inter idx)
  v2f qa[8];
#pragma unroll
  for (int s = 0; s < 8; ++s) {
    qa[s].x = qb[(size_t)(4 * s + koff) * NN + n0 + lm];
    qa[s].y = qb[(size_t)(4 * s + koff + 1) * NN + n0 + lm];
  }

  float mrun[8], lrun[8];
  v8f oacc[4] = {};
#pragma unroll
  for (int i = 0; i < 8; ++i) {
    mrun[i] = -INFINITY;
    lrun[i] = 0.f;
  }

  for (int m0 = 0; m0 < NN; m0 += 16) {
    if (m0 + 16 < NN) {  // prefetch next K / V tiles (global_prefetch_b8)
      __builtin_prefetch(kb + (size_t)koff * NN + m0 + 16 + lm, 0, 1);
      __builtin_prefetch(vb + (size_t)(m0 + 16) * CC + lane, 0, 1);
    }
    // S tile: 16 queries x 16 keys, K = 32
    v8f sa = {};
#pragma unroll
    for (int s = 0; s < 8; ++s) {
      v2f kbv = {kb[(size_t)(4 * s + koff) * NN + m0 + lm],
                 kb[(size_t)(4 * s + koff + 1) * NN + m0 + lm]};
      sa = wmma4(qa[s], kbv, sa);
    }

    // Online softmax (rows spread: vgpr i -> row i + 8*hi, col = lm).
#pragma unroll
    for (int i = 0; i < 8; ++i) {
      float tmax = sa[i];
#pragma unroll
      for (int d = 1; d < 16; d <<= 1)
        tmax = fmaxf(tmax, __shfl_xor(tmax, d, 16));
      float mnew = fmaxf(mrun[i], tmax);
      float corr = __expf(mrun[i] - mnew);
      float p = __expf(sa[i] - mnew);
      float ps = p;
#pragma unroll
      for (int d = 1; d < 16; d <<= 1) ps += __shfl_xor(ps, d, 16);
      lrun[i] = lrun[i] * corr + ps;
      mrun[i] = mnew;
#pragma unroll
      for (int cb2 = 0; cb2 < 4; ++cb2) oacc[cb2][i] *= corr;
      pl[(i + 8 * hi) * 16 + lm] = p;  // P in row-major LDS
    }
    asm volatile("s_wait_dscnt 0" ::: "memory");

    // O += P(16x16) * vT(16x64): A from LDS, B coalesced from vT.
#pragma unroll
    for (int s = 0; s < 4; ++s) {
      v2f pa = *(const v2f*)(pl + lm * 16 + 4 * s + koff);
#pragma unroll
      for (int cb2 = 0; cb2 < 4; ++cb2) {
        v2f bvv = {vb[(size_t)(m0 + 4 * s + koff) * CC + cb2 * 16 + lm],
                   vb[(size_t)(m0 + 4 * s + koff + 1) * CC + cb2 * 16 + lm]};
        oacc[cb2] = wmma4(pa, bvv, oacc[cb2]);
      }
    }
    asm volatile("s_wait_dscnt 0" ::: "memory");  // loads done before next stores
  }

  // Normalize and store O transposed layout outpT[b][n][c] (coalesced in c).
  float* dst = outpT + (size_t)b * NN * CC;
#pragma unroll
  for (int i = 0; i < 8; ++i) {
    float inv = 1.f / lrun[i];
    int row = n0 + i + 8 * hi;
#pragma unroll
    for (int cb2 = 0; cb2 < 4; ++cb2)
      dst[(size_t)row * CC + cb2 * 16 + lm] = oacc[cb2][i] * inv;
  }
}

// ---------------------------------------------------------------------------
// Kernel 6: out = 2*x + gamma_c*out_c + gamma_p*out_p
// ---------------------------------------------------------------------------
__global__ void combine_kernel(const float* __restrict__ x,
                               const float* __restrict__ outc,
                               const float* __restrict__ outpT,
                               const float* __restrict__ gamma_c,
                               const float* __restrict__ gamma_p,
                               float* __restrict__ out) {
  int idx = blockIdx.x * blockDim.x + threadIdx.x;
  if (idx >= BB * CC * NN) return;
  int b = idx / (CC * NN);
  int r = idx - b * (CC * NN);
  int c = r / NN;
  int n = r - c * NN;
  float gc = gamma_c[0], gp = gamma_p[0];
  out[idx] = 2.f * x[idx] + gc * outc[idx] +
             gp * outpT[(size_t)b * NN * CC + (size_t)n * CC + c];
}

// ---------------------------------------------------------------------------
extern "C" void kernel_launch(void* const* d_in, const int* in_sizes, int n_in,
                              void* d_out, int out_size, void* d_ws,
                              size_t ws_size, hipStream_t stream) {
  const float* x = (const float*)d_in[0];
  const float* Wq = (const float*)d_in[1];
  const float* bq = (const float*)d_in[2];
  const float* Wk = (const float*)d_in[3];
  const float* bk = (const float*)d_in[4];
  const float* Wv = (const float*)d_in[5];
  const float* bv = (const float*)d_in[6];
  const float* gamma_c = (const float*)d_in[7];
  const float* gamma_p = (const float*)d_in[8];
  float* out = (float*)d_out;

  float* ws = (float*)d_ws;
  float* q = ws;                                  // B*II*N = 512000
  float* k = q + (size_t)BB * II * NN;            // 512000
  float* vT = k + (size_t)BB * II * NN;           // B*N*C = 1024000
  float* ec = vT + (size_t)BB * NN * CC;          // B*64*64 = 8192
  float* outc = ec + (size_t)BB * CC * CC;        // 1024000
  float* outpT = outc + (size_t)BB * CC * NN;     // 1024000

  qkv_kernel<<<(BB * NN + 127) / 128, 128, 0, stream>>>(x, Wq, bq, Wk, bk, Wv,
                                                        bv, q, k, vT);
  energy_c_kernel<<<BB * 16, 32, 0, stream>>>(x, ec);
  softmax_c_kernel<<<1, 128, 0, stream>>>(ec);
  out_c_kernel<<<BB * 4 * (NN / 16), 32, 0, stream>>>(x, ec, outc);
  pos_attn_kernel<<<BB * (NN / 16), 32, 0, stream>>>(q, k, vT, outpT);
  combine_kernel<<<(BB * CC * NN + 255) / 256, 256, 0, stream>>>(
      x, outc, outpT, gamma_c, gamma_p, out);
}